// Attention__89429809037710
// MI455X (gfx1250) — compile-verified
//
#include <hip/hip_runtime.h>
#include <hip/hip_bf16.h>

// Performer (FAVOR+) causal linear attention forward for MI455X (gfx1250).
// fp32 throughout via V_WMMA_F32_16X16X4_F32 (wave32 WMMA).
//
// Workspace layout (floats), total ~137 MB:
//   q,k,v   : 3 x 2*8*2048*64   = 3 x 2,097,152
//   qf,kf   : 2 x 2*8*2048*256  = 2 x 8,388,608
//   sloc    : 16*16*256*80      = 5,242,880   (per-chunk local [S|Z|..] ext sums)
//   spref   : 5,242,880                         (exclusive prefixes + EPS col)
//   attn    : 2*2048*512        = 2,097,152
//   kmax    : 16

typedef __attribute__((ext_vector_type(2))) float v2f;
typedef __attribute__((ext_vector_type(8))) float v8f;

__device__ __forceinline__ v8f z8() {
  v8f r;
  for (int i = 0; i < 8; ++i) r[i] = 0.0f;
  return r;
}

__device__ __forceinline__ v8f wmma4(v2f a, v2f b, v8f c) {
  // D = A(16x4,f32) * B(4x16,f32) + C(16x16,f32)
  return __builtin_amdgcn_wmma_f32_16x16x4_f32(
      /*neg_a=*/false, a, /*neg_b=*/false, b,
      /*c_mod=*/(short)0, c, /*reuse_a=*/false, /*reuse_b=*/false);
}

// A fragment from row-major [M][K] tile (ld = row stride in floats).
// ISA 16x4 f32 A layout: lanes 0-15 M=0..15 K={k0,k0+1}; lanes 16-31 K={k0+2,k0+3}.
__device__ __forceinline__ v2f frag_a(const float* S, int ld, int m0, int k0, int lane) {
  const float* p = S + (size_t)(m0 + (lane & 15)) * ld + (k0 + ((lane >> 4) << 1));
  v2f r;
  r.x = p[0];
  r.y = p[1];
  return r;
}

// B fragment from row-major [K][N] tile.
// 4x16 f32 B layout (row-striped like C): VGPR j -> K=k0+j (lanes 0-15), K=k0+2+j (lanes 16-31).
__device__ __forceinline__ v2f frag_b(const float* S, int ld, int k0, int n0, int lane) {
  const float* p = S + (size_t)(k0 + ((lane >> 4) << 1)) * ld + (n0 + (lane & 15));
  v2f r;
  r.x = p[0];
  r.y = p[ld];
  return r;
}

// ---------------------------------------------------------------------------
// Generic tiled GEMM: C = alpha*A @ B (+bias), 128x256 block tile, TK=32,
// 8 waves arranged 2(m) x 4(n); each wave owns a 64x64 register tile
// (4x4 subtiles, 16 v8f accumulators) -> 16 WMMAs per 8 fragment loads,
// cutting LDS traffic per WMMA by a third vs a 2x4 tile (f32 WMMA is LDS-bw bound).
// mode 0: plain row-major out. mode 1: QKV head-split out. mode 2: +bias.
// b_transposed: B stored [N][K] row-major (compute A @ B^T).
// ---------------------------------------------------------------------------
__global__ __launch_bounds__(256) void gemm_kernel(
    const float* __restrict__ A, const float* __restrict__ B,
    const float* __restrict__ bias, float* __restrict__ C,
    int M, int Ncols, int K, float alpha, int b_transposed, int mode) {
  __shared__ float As[128 * 33];  // 16.9 KB
  __shared__ float Bs[32 * 260];  // 33.3 KB
  const int tid = threadIdx.x;
  const int lane = tid & 31;
  const int wave = tid >> 5;
  const int wm = wave & 1;   // 2 row groups of 64 rows
  const int wn = wave >> 1;  // 4 col groups of 64 cols
  const int m0 = blockIdx.x * 128;
  const int n0 = blockIdx.y * 256;

  v8f acc[4][4];
  for (int i = 0; i < 4; ++i)
    for (int j = 0; j < 4; ++j) acc[i][j] = z8();

  for (int kt = 0; kt < K; kt += 32) {
    for (int idx = tid; idx < 128 * 32; idx += 256) {
      int kl = idx & 31, r = idx >> 5;
      As[r * 33 + kl] = alpha * A[(size_t)(m0 + r) * K + (kt + kl)];
    }
    if (b_transposed) {
      for (int idx = tid; idx < 32 * 256; idx += 256) {
        int kl = idx & 31, c = idx >> 5;
        Bs[kl * 260 + c] = B[(size_t)(n0 + c) * K + (kt + kl)];
      }
    } else {
      for (int idx = tid; idx < 32 * 256; idx += 256) {
        int c = idx & 255, kl = idx >> 8;
        Bs[kl * 260 + c] = B[(size_t)(kt + kl) * Ncols + (n0 + c)];
      }
    }
    // Prefetch next k-tile into cache while this one computes
    // (lowered to global_prefetch_b8 on gfx1250).
    if (kt + 32 < K) {
      if (tid < 128) __builtin_prefetch(&A[(size_t)(m0 + tid) * K + (kt + 32)], 0, 1);
      if (b_transposed)
        __builtin_prefetch(&B[(size_t)(n0 + tid) * K + (kt + 32)], 0, 1);
      else
        __builtin_prefetch(&B[(size_t)(kt + 32 + (tid >> 3)) * Ncols + n0 + ((tid & 7) << 5)], 0, 1);
    }
    __syncthreads();
    for (int kk = 0; kk < 32; kk += 4) {
      v2f af[4], bf[4];
      for (int i = 0; i < 4; ++i) af[i] = frag_a(As, 33, wm * 64 + i * 16, kk, lane);
      for (int j = 0; j < 4; ++j) bf[j] = frag_b(Bs, 260, kk, wn * 64 + j * 16, lane);
      for (int i = 0; i < 4; ++i)
        for (int j = 0; j < 4; ++j) acc[i][j] = wmma4(af[i], bf[j], acc[i][j]);
    }
    __syncthreads();
  }

  for (int i = 0; i < 4; ++i)
    for (int j = 0; j < 4; ++j) {
      int mb = m0 + wm * 64 + i * 16 + ((lane >> 4) << 3);
      int nn = n0 + wn * 64 + j * 16 + (lane & 15);
      for (int e = 0; e < 8; ++e) {
        int mm = mb + e;
        float val = acc[i][j][e];
        if (mode == 2) val += bias[nn];
        if (mode == 1) {
          // rows are [b*2048 + n], cols are [h*64 + d] -> out[b][h][n][d]
          int bi = mm >> 11, n = mm & 2047, h = nn >> 6, d = nn & 63;
          C[((((size_t)bi * 8 + h) * 2048 + n) << 6) + d] = val;
        } else {
          C[(size_t)mm * Ncols + nn] = val;
        }
      }
    }
}

// ---------------------------------------------------------------------------
// Per-head global max of raw k features (for is_query=False stop-grad max).
// ---------------------------------------------------------------------------
__global__ __launch_bounds__(256) void kmax_kernel(const float* __restrict__ kfraw,
                                                   float* __restrict__ kmax) {
  __shared__ float red[256];
  const int bh = blockIdx.x;
  const float* p = kfraw + (size_t)bh * 2048 * 256;
  float mx = -3.4e38f;
  for (int i = threadIdx.x; i < 2048 * 256; i += 256) mx = fmaxf(mx, p[i]);
  red[threadIdx.x] = mx;
  __syncthreads();
  for (int s = 128; s > 0; s >>= 1) {
    if (threadIdx.x < s) red[threadIdx.x] = fmaxf(red[threadIdx.x], red[threadIdx.x + s]);
    __syncthreads();
  }
  if (threadIdx.x == 0) kmax[bh] = red[0];
}

// ---------------------------------------------------------------------------
// FAVOR+ exp kernel: df = ratio*(exp(df - diag - m) + 1e-4), in place.
// One wave per row of 256 features; diag from raw q/k row (64 floats).
// ---------------------------------------------------------------------------
__global__ __launch_bounds__(256) void feat_exp_kernel(float* __restrict__ df,
                                                       const float* __restrict__ g,
                                                       const float* __restrict__ kmax,
                                                       int is_query) {
  const int row = blockIdx.x * 8 + (threadIdx.x >> 5);
  const int lane = threadIdx.x & 31;
  // diag = 0.5 * (64^-0.5) * sum(x^2) = 0.0625 * sumsq
  float s = 0.0f;
  for (int i = 0; i < 2; ++i) {
    float x = g[(size_t)row * 64 + lane + 32 * i];
    s += x * x;
  }
  for (int m = 16; m > 0; m >>= 1) s += __shfl_xor(s, m, 32);
  const float diag = 0.0625f * s;

  float vals[8];
  float mx = -3.4e38f;
  for (int i = 0; i < 8; ++i) {
    vals[i] = df[(size_t)row * 256 + lane + 32 * i];
    mx = fmaxf(mx, vals[i]);
  }
  if (is_query) {
    for (int m = 16; m > 0; m >>= 1) mx = fmaxf(mx, __shfl_xor(mx, m, 32));
  } else {
    mx = kmax[row >> 11];  // per (b,h)
  }
  for (int i = 0; i < 8; ++i)
    df[(size_t)row * 256 + lane + 32 * i] = 0.0625f * (expf(vals[i] - diag - mx) + 1e-4f);
}

// ---------------------------------------------------------------------------
// Per-chunk local sums: S_loc_ext[256x80] = kc^T @ [vc | 1 | 0...].
// Col 64 = column-sum of kc (local Z). Block per (chunk, head).
// ---------------------------------------------------------------------------
__global__ __launch_bounds__(256) void local_sums_kernel(const float* __restrict__ kf,
                                                         const float* __restrict__ v,
                                                         float* __restrict__ sloc) {
  __shared__ float Ks[256 * 33];  // [f][pos]
  __shared__ float Vs[32 * 84];   // [pos][col]
  const int c = blockIdx.x, bh = blockIdx.y;
  const int tid = threadIdx.x, lane = tid & 31, wave = tid >> 5;
  const size_t rbase = (size_t)bh * 2048 + c * 128;

  v8f acc[2][5];
  for (int i = 0; i < 2; ++i)
    for (int j = 0; j < 5; ++j) acc[i][j] = z8();

  for (int pt = 0; pt < 128; pt += 32) {
    for (int idx = tid; idx < 32 * 256; idx += 256) {
      int f = idx & 255, pos = idx >> 8;
      Ks[f * 33 + pos] = kf[(rbase + pt + pos) * 256 + f];  // transpose: [f][pos]
    }
    for (int idx = tid; idx < 32 * 80; idx += 256) {
      int col = idx % 80, pos = idx / 80;
      float val = (col < 64) ? v[(rbase + pt + pos) * 64 + col] : (col == 64 ? 1.0f : 0.0f);
      Vs[pos * 84 + col] = val;
    }
    __syncthreads();
    for (int kk = 0; kk < 32; kk += 4) {
      v2f af[2], bf[5];
      for (int i = 0; i < 2; ++i) af[i] = frag_a(Ks, 33, wave * 32 + i * 16, kk, lane);
      for (int j = 0; j < 5; ++j) bf[j] = frag_b(Vs, 84, kk, j * 16, lane);
      for (int i = 0; i < 2; ++i)
        for (int j = 0; j < 5; ++j) acc[i][j] = wmma4(af[i], bf[j], acc[i][j]);
    }
    __syncthreads();
  }

  float* out = sloc + (size_t)(bh * 16 + c) * 256 * 80;
  for (int i = 0; i < 2; ++i)
    for (int j = 0; j < 5; ++j) {
      int mb = wave * 32 + i * 16 + ((lane >> 4) << 3);
      int nb = j * 16 + (lane & 15);
      for (int e = 0; e < 8; ++e) out[(size_t)(mb + e) * 80 + nb] = acc[i][j][e];
    }
}

// ---------------------------------------------------------------------------
// Exclusive prefix over chunks; also emits Z,S outputs (inclusive totals).
// Col 65 of spref is forced to EPS (folds EPS*rowsum(q) into the GEMM).
// ---------------------------------------------------------------------------
__global__ void prefix_kernel(const float* __restrict__ sloc, float* __restrict__ spref,
                              float* __restrict__ Zout, float* __restrict__ Sout) {
  const int idx = blockIdx.x * blockDim.x + threadIdx.x;
  if (idx >= 16 * 256 * 80) return;
  const int col = idx % 80;
  const int f = (idx / 80) % 256;
  const int bh = idx / (80 * 256);
  float acc = 0.0f;
  for (int c = 0; c < 16; ++c) {
    size_t o = ((size_t)(bh * 16 + c) * 256 + f) * 80 + col;
    float pref;
    if (col == 65) pref = 1e-6f;      // EPS column
    else if (col > 65) pref = 0.0f;
    else pref = acc;                  // exclusive prefix (cols 0..64)
    spref[o] = pref;
    acc += sloc[o];
  }
  if (col < 64) Sout[((size_t)bh * 256 + f) * 64 + col] = acc;
  else if (col == 64) Zout[(size_t)bh * 256 + f] = acc;
}

// ---------------------------------------------------------------------------
// Chunk attention: per (chunk, head) block, two 64-row halves:
//   A = qc @ kc^T (tril-masked, inclusive)
//   NUM = qc @ S_ext + tril(A) @ V_ext        (N=80; col64+col65 = denominator)
//   attn = NUM[:, :64] / (NUM[:,64] + NUM[:,65])
// ---------------------------------------------------------------------------
__global__ __launch_bounds__(256) void chunk_attn_kernel(
    const float* __restrict__ qf, const float* __restrict__ kf,
    const float* __restrict__ v, const float* __restrict__ spref,
    float* __restrict__ attn) {
  __shared__ float As2[64 * 132];  // masked A half-tile, later reused as NUM[64][84]
  __shared__ float Qs[64 * 33];
  __shared__ float Bs[32 * 132];
  const int c = blockIdx.x, bh = blockIdx.y;
  const int tid = threadIdx.x, lane = tid & 31, wave = tid >> 5;
  const size_t rbase = (size_t)bh * 2048 + c * 128;
  const float* sp = spref + (size_t)(bh * 16 + c) * 256 * 80;
  const int b = bh >> 3, h = bh & 7;

  for (int rb = 0; rb < 2; ++rb) {
    const size_t rowg = rbase + rb * 64;
    // ---- Phase 1: A = qc[half] @ kc^T  (M=64, N=128, K=256)
    const int wm = wave & 3;   // 4 m-subtiles of 16 rows
    const int wn = wave >> 2;  // 2 col groups of 64
    v8f accA[4];
    for (int j = 0; j < 4; ++j) accA[j] = z8();
    for (int kt = 0; kt < 256; kt += 32) {
      for (int idx = tid; idx < 64 * 32; idx += 256) {
        int kl = idx & 31, m = idx >> 5;
        Qs[m * 33 + kl] = qf[(rowg + m) * 256 + (kt + kl)];
      }
      for (int idx = tid; idx < 128 * 32; idx += 256) {
        int kl = idx & 31, n = idx >> 5;
        Bs[kl * 132 + n] = kf[(rbase + n) * 256 + (kt + kl)];  // kc^T: [f][pos]
      }
      __syncthreads();
      for (int kk = 0; kk < 32; kk += 4) {
        v2f a = frag_a(Qs, 33, wm * 16, kk, lane);
        for (int j = 0; j < 4; ++j) {
          v2f bfr = frag_b(Bs, 132, kk, wn * 64 + j * 16, lane);
          accA[j] = wmma4(a, bfr, accA[j]);
        }
      }
      __syncthreads();
    }
    // ---- Phase 2: tril mask (col <= global row) -> LDS
    {
      int mb = wm * 16 + ((lane >> 4) << 3);
      for (int j = 0; j < 4; ++j) {
        int nb = wn * 64 + j * 16 + (lane & 15);
        for (int e = 0; e < 8; ++e) {
          int mm = mb + e;
          As2[mm * 132 + nb] = (nb <= rb * 64 + mm) ? accA[j][e] : 0.0f;
        }
      }
    }
    __syncthreads();
    // ---- Phase 3: NUM = qc @ S_ext + trilA @ V_ext (M=64, N=80)
    v8f accN[3];
    for (int j = 0; j < 3; ++j) accN[j] = z8();
    const int wrow = (wave & 3) * 16;
    const int nsub0 = (wave >> 2) ? 3 : 0;
    const int nsubc = (wave >> 2) ? 2 : 3;
    // 3a: qc @ S_ext, K=256
    for (int kt = 0; kt < 256; kt += 32) {
      for (int idx = tid; idx < 64 * 32; idx += 256) {
        int kl = idx & 31, m = idx >> 5;
        Qs[m * 33 + kl] = qf[(rowg + m) * 256 + (kt + kl)];
      }
      for (int idx = tid; idx < 32 * 80; idx += 256) {
        int col = idx % 80, kl = idx / 80;
        Bs[kl * 132 + col] = sp[(size_t)(kt + kl) * 80 + col];
      }
      __syncthreads();
      for (int kk = 0; kk < 32; kk += 4) {
        v2f a = frag_a(Qs, 33, wrow, kk, lane);
        for (int j = 0; j < nsubc; ++j) {
          v2f bfr = frag_b(Bs, 132, kk, (nsub0 + j) * 16, lane);
          accN[j] = wmma4(a, bfr, accN[j]);
        }
      }
      __syncthreads();
    }
    // 3b: += trilA @ V_ext, K=128 chunk positions
    for (int kt = 0; kt < 128; kt += 32) {
      for (int idx = tid; idx < 32 * 80; idx += 256) {
        int col = idx % 80, kl = idx / 80;
        float val = (col < 64) ? v[(rbase + kt + kl) * 64 + col] : (col == 64 ? 1.0f : 0.0f);
        Bs[kl * 132 + col] = val;
      }
      __syncthreads();
      for (int kk = 0; kk < 32; kk += 4) {
        v2f a = frag_a(As2, 132, wrow, kt + kk, lane);
        for (int j = 0; j < nsubc; ++j) {
          v2f bfr = frag_b(Bs, 132, kk, (nsub0 + j) * 16, lane);
          accN[j] = wmma4(a, bfr, accN[j]);
        }
      }
      __syncthreads();
    }
    // ---- Phase 4: NUM -> LDS (reuse As2 as [64][84]); normalize; write attn
    {
      int mb = wrow + ((lane >> 4) << 3);
      for (int j = 0; j < nsubc; ++j) {
        int nb = (nsub0 + j) * 16 + (lane & 15);
        for (int e = 0; e < 8; ++e) As2[(mb + e) * 84 + nb] = accN[j][e];
      }
    }
    __syncthreads();
    {
      int r = tid >> 2;
      int e0 = (tid & 3) * 16;
      float D = As2[r * 84 + 64] + As2[r * 84 + 65];
      float rinv = 1.0f / D;
      size_t orow = ((size_t)b * 2048 + c * 128 + rb * 64 + r) * 512 + h * 64;
      for (int e = 0; e < 16; ++e) attn[orow + e0 + e] = As2[r * 84 + e0 + e] * rinv;
    }
    __syncthreads();
  }
}

// ---------------------------------------------------------------------------
extern "C" void kernel_launch(void* const* d_in, const int* in_sizes, int n_in,
                              void* d_out, int out_size, void* d_ws, size_t ws_size,
                              hipStream_t stream) {
  const float* x = (const float*)d_in[0];     // [2,2048,512]
  const float* proj = (const float*)d_in[1];  // [256,64]
  const float* Wq = (const float*)d_in[2];
  const float* Wk = (const float*)d_in[3];
  const float* Wv = (const float*)d_in[4];
  const float* Wo = (const float*)d_in[5];
  const float* bo = (const float*)d_in[6];

  float* ws = (float*)d_ws;
  float* q = ws;
  float* k = q + 2097152;
  float* v = k + 2097152;
  float* qf = v + 2097152;
  float* kf = qf + 8388608;
  float* sloc = kf + 8388608;
  float* spref = sloc + 5242880;
  float* attn = spref + 5242880;
  float* kmaxb = attn + 2097152;

  float* out = (float*)d_out;           // [2,2048,512]
  float* Zout = out + 2097152;          // [2,8,256]
  float* Sout = Zout + 4096;            // [2,8,256,64]

  const float normalizer = 0.35355339059327373f;  // 64^-0.25

  // 1) QKV projections (head-split output layout)
  dim3 g1(32, 2);  // 128x256 tiles over [4096 x 512]
  gemm_kernel<<<g1, 256, 0, stream>>>(x, Wq, nullptr, q, 4096, 512, 512, 1.0f, 0, 1);
  gemm_kernel<<<g1, 256, 0, stream>>>(x, Wk, nullptr, k, 4096, 512, 512, 1.0f, 0, 1);
  gemm_kernel<<<g1, 256, 0, stream>>>(x, Wv, nullptr, v, 4096, 512, 512, 1.0f, 0, 1);

  // 2) Feature GEMMs: data_dash = (g * normalizer) @ proj^T
  dim3 g2(256, 1);  // 128x256 tiles over [32768 x 256]
  gemm_kernel<<<g2, 256, 0, stream>>>(q, proj, nullptr, qf, 32768, 256, 64, normalizer, 1, 0);
  gemm_kernel<<<g2, 256, 0, stream>>>(k, proj, nullptr, kf, 32768, 256, 64, normalizer, 1, 0);

  // 3) Per-head max for k, then FAVOR+ exp (in place)
  kmax_kernel<<<16, 256, 0, stream>>>(kf, kmaxb);
  feat_exp_kernel<<<4096, 256, 0, stream>>>(qf, q, nullptr, 1);
  feat_exp_kernel<<<4096, 256, 0, stream>>>(kf, k, kmaxb, 0);

  // 4) Chunk-local state sums, prefix scan (also emits Z,S outputs)
  dim3 g5(16, 16);  // (chunk, head)
  local_sums_kernel<<<g5, 256, 0, stream>>>(kf, v, sloc);
  prefix_kernel<<<1280, 256, 0, stream>>>(sloc, spref, Zout, Sout);

  // 5) Chunk-parallel causal linear attention
  chunk_attn_kernel<<<g5, 256, 0, stream>>>(qf, kf, v, spref, attn);

  // 6) Output projection + bias
  dim3 g6(32, 2);
  gemm_kernel<<<g6, 256, 0, stream>>>(attn, Wo, bo, out, 4096, 512, 512, 1.0f, 0, 2);
}